// WeightSpaceModel_45698452029678
// MI455X (gfx1250) — compile-verified
//
#include <hip/hip_runtime.h>
#include <hip/hip_bf16.h>

// ---------------- problem constants ----------------
#define PDIM   4354          // total flat parameter count
#define PDIM4  4356          // PDIM rounded up to multiple of 4 (finalize vector width)
#define KPAD   4480          // K padded: 8 segs x 140 chunks x 4 = clean trip counts
#define NPAD   4368          // N padded to multiple of 16 (WMMA n-tile)
#define BATCH  32
#define SEQ    64
#define NTILES 273           // ceil(PDIM/16)
#define KSEG   8             // K-split factor for parallelism
#define SEGF   (KPAD / KSEG) // 560 floats of K per segment

typedef __attribute__((ext_vector_type(2))) float v2f;
typedef __attribute__((ext_vector_type(8))) float v8f;

// ---------------- one-time: transpose + pad A into pair-interleaved Atp ----------
// Atp pair-row r, column n holds float2 { A.T[2r][n], A.T[2r+1][n] } so a WMMA
// B-fragment is a single b64 load per lane.
__global__ __launch_bounds__(256) void transpose_pad_kernel(
    const float* __restrict__ A, float* __restrict__ Atp) {
  __shared__ float tile[16][17];
  const int kb = blockIdx.x * 16;
  const int nb = blockIdx.y * 16;
  const int tx = threadIdx.x, ty = threadIdx.y;
  // coalesced read: A[n][k], k fastest
  const int nr = nb + ty, kr = kb + tx;
  float v = 0.0f;
  if (nr < PDIM && kr < PDIM) v = A[(size_t)nr * PDIM + kr];
  tile[ty][tx] = v;
  __syncthreads();
  // write pair-interleaved: element (k, n) -> Atp[(k>>1)*2*NPAD + 2*n + (k&1)]
  const int kw = kb + ty, nw = nb + tx;
  if (kw < KPAD && nw < NPAD)
    Atp[(size_t)(kw >> 1) * (2 * NPAD) + 2 * nw + (kw & 1)] = tile[tx][ty];
}

// ---------------- init theta ping-pong buffers, padded Bvec, x_prev ----------------
__global__ __launch_bounds__(256) void init_kernel(
    const float* __restrict__ theta_init, const float* __restrict__ Bvec,
    float* __restrict__ thetaA, float* __restrict__ thetaB,
    float* __restrict__ Bpad, float* __restrict__ x_prev) {
  const int idx = blockIdx.x * blockDim.x + threadIdx.x;
  if (idx < BATCH * KPAD) {
    const int k = idx % KPAD;
    thetaA[idx] = (k < PDIM) ? theta_init[k] : 0.0f;
    thetaB[idx] = 0.0f;                    // pad columns must stay zero
  }
  if (idx < PDIM4) Bpad[idx] = (idx < PDIM) ? Bvec[idx] : 0.0f;
  if (idx < BATCH) x_prev[idx] = 0.0f;
}

// ---------------- per-step head: outputs, x_t, u = x_t - x_prev ----------------
__global__ __launch_bounds__(32) void step_head_kernel(
    const float* __restrict__ mu_sigma, const float* __restrict__ xs,
    const float* __restrict__ noise, const int* __restrict__ inf_start,
    float* __restrict__ x_prev, float* __restrict__ u,
    float* __restrict__ out, int t) {
  const int b = threadIdx.x;
  const float mean = mu_sigma[b * 2 + 0];
  const float std  = mu_sigma[b * 2 + 1];
  out[b * SEQ + t]               = mean;   // means.T -> (batch, seq)
  out[BATCH * SEQ + b * SEQ + t] = std;    // stds.T
  const float xhat = noise[t * BATCH + b] * std + mean;
  const float xt = (t < *inf_start) ? xs[b * SEQ + t] : xhat;
  u[b] = xt - x_prev[b];
  x_prev[b] = xt;
}

// ---------------- GEMM partials: theta_part[s] = theta_in @ A.T (K-segment s)
// One wave per (16-column n-tile, K-segment); full M=32 via two accumulators.
__global__ __launch_bounds__(256) void gemm_step_kernel(
    const float* __restrict__ theta_in, float* __restrict__ theta_part,
    const float* __restrict__ Atp) {
  const int lane = threadIdx.x & 31;
  const int wave = threadIdx.x >> 5;
  const int tile = blockIdx.x * 8 + wave;
  if (tile >= NTILES) return;              // wave-uniform exit; WMMA waves keep EXEC=all-1s
  const int kseg = blockIdx.y;
  const int k0   = kseg * SEGF;

  const int half = lane >> 4;              // 0: k offsets {0,1}; 1: {2,3}
  const int l16  = lane & 15;
  const int n    = tile * 16 + l16;
  const int kb   = half * 2;

  v8f c0 = {};                             // batch rows 0..15
  v8f c1 = {};                             // batch rows 16..31
  const float* pa0 = theta_in + (size_t)l16 * KPAD + k0 + kb;
  const float* pa1 = theta_in + (size_t)(16 + l16) * KPAD + k0 + kb;
  const float* pb  = Atp + (size_t)(k0 / 2 + half) * (2 * NPAD) + 2 * n;

  // SEGF/16 = 35 bodies, each covering 16 k via 4 WMMA chunk pairs; no remainder.
  for (int o = 0; o < SEGF / 16; ++o) {
    __builtin_prefetch(pb + (size_t)16 * (4 * NPAD), 0, 1);  // B stream, 64 k ahead
#pragma unroll
    for (int i = 0; i < 4; ++i) {
      const v2f a0 = *(const v2f*)(pa0 + 4 * i);
      const v2f a1 = *(const v2f*)(pa1 + 4 * i);
      const v2f b  = *(const v2f*)(pb + (size_t)i * (4 * NPAD));
      c0 = __builtin_amdgcn_wmma_f32_16x16x4_f32(false, a0, false, b,
                                                 (short)0, c0, false, false);
      c1 = __builtin_amdgcn_wmma_f32_16x16x4_f32(false, a1, false, b,
                                                 (short)0, c1, false, false);
    }
    pa0 += 16; pa1 += 16; pb += (size_t)4 * (4 * NPAD);
  }

  if (n < PDIM4) {                         // cols [PDIM, PDIM4) produce exact zeros
    float* po = theta_part + (size_t)kseg * (BATCH * KPAD) + n;
#pragma unroll
    for (int r = 0; r < 8; ++r) {
      const int m0 = r + 8 * half;         // C/D layout: VGPR r -> row m0 (+16 other half)
      po[(size_t)m0 * KPAD]        = c0[r];
      po[(size_t)(m0 + 16) * KPAD] = c1[r];
    }
  }
}

// ---------------- finalize: theta_out = clip(sum_s part[s] + u*Bpad, -1, 1) ------
__global__ __launch_bounds__(256) void finalize_kernel(
    const float* __restrict__ theta_part, const float* __restrict__ Bpad,
    const float* __restrict__ u, float* __restrict__ theta_out) {
  const int nq  = PDIM4 / 4;               // 1089 float4s per batch row
  const int idx = blockIdx.x * blockDim.x + threadIdx.x;
  if (idx >= BATCH * nq) return;
  const int m  = idx / nq;
  const int n4 = (idx % nq) * 4;

  float4 acc = make_float4(0.f, 0.f, 0.f, 0.f);
#pragma unroll
  for (int s = 0; s < KSEG; ++s) {
    const float4 p = *(const float4*)(theta_part + (size_t)s * (BATCH * KPAD) +
                                      (size_t)m * KPAD + n4);
    acc.x += p.x; acc.y += p.y; acc.z += p.z; acc.w += p.w;
  }
  const float um = u[m];
  const float4 bv = *(const float4*)(Bpad + n4);
  float4 o;
  o.x = fminf(fmaxf(acc.x + um * bv.x, -1.0f), 1.0f);
  o.y = fminf(fmaxf(acc.y + um * bv.y, -1.0f), 1.0f);
  o.z = fminf(fmaxf(acc.z + um * bv.z, -1.0f), 1.0f);
  o.w = fminf(fmaxf(acc.w + um * bv.w, -1.0f), 1.0f);
  *(float4*)(theta_out + (size_t)m * KPAD + n4) = o;
}

// ---------------- hypernetwork MLP: mu_sigma[b] = mlp(theta[b], t_in) ----------------
__global__ __launch_bounds__(32) void mlp_kernel(
    const float* __restrict__ theta, float t_in,
    float* __restrict__ mu_sigma) {
  const int b = blockIdx.x;
  const int j = threadIdx.x;               // neuron index
  const float* th = theta + (size_t)b * KPAD;
  __shared__ float y[32];

  // layer 0: (1 -> 32), W at [0,32), bias at [32,64)
  float z = th[j] * t_in + th[32 + j];
  float h = z / (1.0f + __expf(-z));       // silu

  int off = 64;
#pragma unroll
  for (int l = 0; l < 4; ++l) {            // 4 hidden (32 -> 32) layers
    y[j] = h;
    __syncthreads();
    float acc = th[off + 1024 + j];        // bias
    const float* Wr = th + off + j * 32;   // row j of W (o,i)
#pragma unroll
    for (int i = 0; i < 32; ++i) acc += Wr[i] * y[i];
    h = acc / (1.0f + __expf(-acc));
    __syncthreads();
    off += 1056;
  }
  y[j] = h;
  __syncthreads();

  if (j < 2) {                             // output layer: (32 -> 2), W at 4288, bias at 4352
    float acc = th[4352 + j];
    const float* Wr = th + 4288 + j * 32;
#pragma unroll
    for (int i = 0; i < 32; ++i) acc += Wr[i] * y[i];
    float o;
    if (j == 0) {
      o = tanhf(acc);                                      // mean
    } else {
      o = __expf(0.5f * fminf(fmaxf(acc, -4.0f), 4.0f));   // std
    }
    mu_sigma[b * 2 + j] = o;
  }
}

// ---------------- host orchestration ----------------
extern "C" void kernel_launch(void* const* d_in, const int* in_sizes, int n_in,
                              void* d_out, int out_size, void* d_ws, size_t ws_size,
                              hipStream_t stream) {
  const float* xs         = (const float*)d_in[0]; // (32, 64)
  const float* noise      = (const float*)d_in[1]; // (64, 32, 1)
  const float* theta_init = (const float*)d_in[2]; // (P,)
  const float* Amat       = (const float*)d_in[3]; // (P, P)
  const float* Bvec       = (const float*)d_in[4]; // (P, 1)
  const int*   inf_start  = (const int*)d_in[5];   // scalar

  float* ws         = (float*)d_ws;
  float* Atp        = ws;                                    // KPAD * NPAD
  float* thetaA     = Atp + (size_t)KPAD * NPAD;             // BATCH * KPAD
  float* thetaB     = thetaA + (size_t)BATCH * KPAD;         // BATCH * KPAD
  float* theta_part = thetaB + (size_t)BATCH * KPAD;         // KSEG * BATCH * KPAD
  float* Bpad       = theta_part + (size_t)KSEG * BATCH * KPAD; // PDIM4
  float* u          = Bpad + PDIM4;                          // BATCH
  float* x_prev     = u + BATCH;                             // BATCH
  float* mu_sigma   = x_prev + BATCH;                        // BATCH * 2
  float* out        = (float*)d_out;

  // one-time per call: transpose+pad A, init theta/Bpad/x_prev, first MLP eval
  transpose_pad_kernel<<<dim3(KPAD / 16, NPAD / 16), dim3(16, 16), 0, stream>>>(Amat, Atp);
  init_kernel<<<(BATCH * KPAD + 255) / 256, 256, 0, stream>>>(
      theta_init, Bvec, thetaA, thetaB, Bpad, x_prev);
  mlp_kernel<<<BATCH, 32, 0, stream>>>(thetaA, 0.0f, mu_sigma);

  const int fin_threads = BATCH * (PDIM4 / 4);
  float* cur = thetaA;
  float* nxt = thetaB;
  for (int t = 0; t < SEQ; ++t) {
    step_head_kernel<<<1, 32, 0, stream>>>(mu_sigma, xs, noise, inf_start,
                                           x_prev, u, out, t);
    gemm_step_kernel<<<dim3((NTILES + 7) / 8, KSEG), 256, 0, stream>>>(
        cur, theta_part, Atp);
    finalize_kernel<<<(fin_threads + 255) / 256, 256, 0, stream>>>(
        theta_part, Bpad, u, nxt);
    mlp_kernel<<<BATCH, 32, 0, stream>>>(nxt, (float)(t + 1) / (float)SEQ, mu_sigma);
    float* tmp = cur; cur = nxt; nxt = tmp;
  }
}